// ATD_CA_9689446219852
// MI455X (gfx1250) — compile-verified
//
#include <hip/hip_runtime.h>

typedef __attribute__((ext_vector_type(2))) float v2f;
typedef __attribute__((ext_vector_type(4))) float v4f;
typedef __attribute__((ext_vector_type(8))) float v8f;

// ds_swizzle bit-mask mode: offset[14:10]=xor, [9:5]=or, [4:0]=and
#define SWZ(x, imm) __int_as_float(__builtin_amdgcn_ds_swizzle(__float_as_int(x), (imm)))

__device__ __forceinline__ float red_max16(float x) {
  x = fmaxf(x, SWZ(x, 0x041F));
  x = fmaxf(x, SWZ(x, 0x081F));
  x = fmaxf(x, SWZ(x, 0x101F));
  x = fmaxf(x, SWZ(x, 0x201F));
  return x;
}
__device__ __forceinline__ float red_sum16(float x) {
  x += SWZ(x, 0x041F);
  x += SWZ(x, 0x081F);
  x += SWZ(x, 0x101F);
  x += SWZ(x, 0x201F);
  return x;
}

// ---------------- pre-kernel 1: kn = l2norm(td @ Wk^T + bk), pair-swizzled ---
// ws_knp[bi][p][m] (float2) = { knT[2p][m], knT[2p+1][m] }, p<6 (K padded 10->12)
__global__ __launch_bounds__(64) void prek_kn(
    const float* __restrict__ td, const float* __restrict__ wk_w,
    const float* __restrict__ wk_b, float* __restrict__ ws_knp)
{
  const int bi = blockIdx.x;
  const int m  = threadIdx.x;                   // 0..63
  const float* tdr = td + ((size_t)(bi * 64 + m)) * 192;
  float kv[12];
  #pragma unroll
  for (int r = 0; r < 10; ++r) {
    float acc = 0.f;
    for (int cc = 0; cc < 192; ++cc) acc = fmaf(tdr[cc], wk_w[r * 192 + cc], acc);
    kv[r] = acc + wk_b[r];
  }
  kv[10] = 0.f; kv[11] = 0.f;
  float ss = 0.f;
  #pragma unroll
  for (int r = 0; r < 10; ++r) ss += kv[r] * kv[r];
  const float inv = 1.0f / fmaxf(sqrtf(ss), 1e-12f);
  #pragma unroll
  for (int p = 0; p < 6; ++p) {
    float* dst = ws_knp + ((size_t)((bi * 6 + p) * 64 + m)) * 2;
    dst[0] = (2 * p     < 10) ? kv[2 * p]     * inv : 0.f;
    dst[1] = (2 * p + 1 < 10) ? kv[2 * p + 1] * inv : 0.f;
  }
}

// ---------------- pre-kernel 2: v = td @ Wv^T + bv, pair-swizzled ------------
// ws_vp[bi][p][cc] (float2) = { v[2p][cc], v[2p+1][cc] }, p<32, cc<192
__global__ __launch_bounds__(256) void prek_v(
    const float* __restrict__ td, const float* __restrict__ wv_w,
    const float* __restrict__ wv_b, float* __restrict__ ws_vp)
{
  const int bi = blockIdx.x / 12;
  const int sl = blockIdx.x % 12;
  #pragma unroll
  for (int j = 0; j < 4; ++j) {
    const int idx = sl * 1024 + j * 256 + threadIdx.x;   // 0..12287
    const int m = idx / 192, cc = idx % 192;
    const float* tdr = td + ((size_t)(bi * 64 + m)) * 192;
    const float* wr  = wv_w + (size_t)cc * 192;
    float acc = 0.f;
    for (int t = 0; t < 192; ++t) acc = fmaf(tdr[t], wr[t], acc);
    acc += wv_b[cc];
    ws_vp[(((size_t)(bi * 32 + (m >> 1))) * 192 + cc) * 2 + (m & 1)] = acc;
  }
}

// ---------------- main fused kernel ------------------------------------------
__global__ __launch_bounds__(256) void attn_main(
    const float* __restrict__ x, const float* __restrict__ wq_w,
    const float* __restrict__ wq_b, const float* __restrict__ scale,
    const float* __restrict__ ws_knp, const float* __restrict__ ws_vp,
    float* __restrict__ gout, float* __restrict__ gattn)
{
  extern __shared__ float lds[];
  const int bi   = blockIdx.x >> 7;       // 8 batches
  const int tg   = blockIdx.x & 127;      // 128 row-groups per batch
  const int tid  = threadIdx.x;
  const int wid  = tid >> 5;              // wave 0..7
  const int lane = tid & 31;
  const int h    = lane >> 4;             // lane half
  const int c    = lane & 15;             // lane-in-half

  float* vp  = lds;                       // (p*208 + cc)*2 floats, p<32, cc<192
  float* wqp = lds + 13312;               // (p*16 + r)*2, p<96, r<16
  float* knp = lds + 16384;               // (p*64 + m)*2, p<6, m<64
  float* scr = lds + 17152 + wid * (16 * 68);  // per-wave 16x68 scratch

  // -- block-cooperative staging into LDS --
  for (int idx = tid; idx < 32 * 192; idx += 256) {
    const int p = idx / 192, cc = idx - p * 192;
    const v2f s = *(const v2f*)(ws_vp + ((size_t)bi * 6144 + idx) * 2);
    *(v2f*)(vp + (p * 208 + cc) * 2) = s;
  }
  for (int idx = tid; idx < 96 * 16; idx += 256) {
    const int p = idx >> 4, r = idx & 15;
    v2f s = {0.f, 0.f};
    if (r < 10) { s.x = wq_w[r * 192 + 2 * p]; s.y = wq_w[r * 192 + 2 * p + 1]; }
    *(v2f*)(wqp + idx * 2) = s;
  }
  for (int idx = tid; idx < 6 * 64; idx += 256) {
    const v2f s = *(const v2f*)(ws_knp + ((size_t)bi * 384 + idx) * 2);
    *(v2f*)(knp + idx * 2) = s;
  }
  __syncthreads();

  const int r0 = tg * 128 + wid * 16;
  const float* xrow = x + ((size_t)bi * 16384 + r0 + c) * 192;   // A row = c

  // ---- Q projection: C[16x16] = X[16x192] * WqT[192x16] (cols>=10 are 0) ----
  v8f qacc = {};
  for (int ks = 0; ks < 48; ++ks) {
    const v2f af = *(const v2f*)(xrow + ks * 4 + 2 * h);
    const v2f bf = *(const v2f*)(wqp + ((ks * 2 + h) * 16 + c) * 2);
    qacc = __builtin_amdgcn_wmma_f32_16x16x4_f32(false, af, false, bf,
                                                 (short)0, qacc, false, false);
  }
  const float bias = (c < 10) ? wq_b[c] : 0.0f;

  // row L2-normalize and stash qn rows in scratch (cols 0..15, >=10 are 0)
  #pragma unroll
  for (int v = 0; v < 8; ++v) {
    const float q  = qacc[v] + bias;
    const float ss = red_sum16(q * q);
    const float iv = 1.0f / fmaxf(sqrtf(ss), 1e-12f);
    scr[(v + 8 * h) * 68 + c] = q * iv;
  }

  // ---- logits: [16x64] = Qn[16x12] * KnT[12x64] ----
  v2f laf[3];
  #pragma unroll
  for (int ks = 0; ks < 3; ++ks)
    laf[ks] = *(const v2f*)(scr + c * 68 + ks * 4 + 2 * h);

  v8f lacc[4];
  #pragma unroll
  for (int t = 0; t < 4; ++t) {
    v8f a = {};
    #pragma unroll
    for (int ks = 0; ks < 3; ++ks) {
      const v2f bf = *(const v2f*)(knp + ((ks * 2 + h) * 64 + t * 16 + c) * 2);
      a = __builtin_amdgcn_wmma_f32_16x16x4_f32(false, laf[ks], false, bf,
                                                (short)0, a, false, false);
    }
    lacc[t] = a;
  }

  const float sc = 1.0f + fminf(fmaxf(scale[0], 0.0f), 3.0f) * 4.1588831f;

  // ---- softmax over 64 tokens per row; store probs to scratch ----
  #pragma unroll
  for (int v = 0; v < 8; ++v) {
    const float l0 = lacc[0][v] * sc, l1 = lacc[1][v] * sc;
    const float l2 = lacc[2][v] * sc, l3 = lacc[3][v] * sc;
    const float mx = red_max16(fmaxf(fmaxf(l0, l1), fmaxf(l2, l3)));
    const float e0 = __expf(l0 - mx), e1 = __expf(l1 - mx);
    const float e2 = __expf(l2 - mx), e3 = __expf(l3 - mx);
    const float sm = red_sum16(e0 + e1 + e2 + e3);
    const float iv = 1.0f / sm;
    float* rp = scr + (v + 8 * h) * 68;
    rp[c]      = e0 * iv;
    rp[16 + c] = e1 * iv;
    rp[32 + c] = e2 * iv;
    rp[48 + c] = e3 * iv;
  }

  // ---- stream attn tile to global (coalesced b128) ----
  {
    float* ga = gattn + ((size_t)bi * 16384 + r0) * 64;
    #pragma unroll
    for (int i = 0; i < 8; ++i) {
      const int idx = i * 32 + lane;
      const int rr = idx >> 4, q4 = idx & 15;
      const v4f val = *(const v4f*)(scr + rr * 68 + q4 * 4);
      *(v4f*)(ga + rr * 64 + q4 * 4) = val;
    }
  }

  // ---- out = attn[16x64] * V[64x192] ----
  v2f oaf[16];
  #pragma unroll
  for (int ks = 0; ks < 16; ++ks)
    oaf[ks] = *(const v2f*)(scr + c * 68 + ks * 4 + 2 * h);

  float* go = gout + ((size_t)bi * 16384 + r0) * 192;
  for (int ct = 0; ct < 12; ++ct) {
    v8f oacc = {};
    #pragma unroll
    for (int ks = 0; ks < 16; ++ks) {
      const v2f bf = *(const v2f*)(vp + ((ks * 2 + h) * 208 + ct * 16 + c) * 2);
      oacc = __builtin_amdgcn_wmma_f32_16x16x4_f32(false, oaf[ks], false, bf,
                                                   (short)0, oacc, false, false);
    }
    #pragma unroll
    for (int v = 0; v < 8; ++v)
      go[(v + 8 * h) * 192 + ct * 16 + c] = oacc[v];
  }
}

extern "C" void kernel_launch(void* const* d_in, const int* in_sizes, int n_in,
                              void* d_out, int out_size, void* d_ws, size_t ws_size,
                              hipStream_t stream) {
  const float* x     = (const float*)d_in[0];
  const float* td    = (const float*)d_in[1];
  const float* wq_w  = (const float*)d_in[2];
  const float* wq_b  = (const float*)d_in[3];
  const float* wk_w  = (const float*)d_in[4];
  const float* wk_b  = (const float*)d_in[5];
  const float* wv_w  = (const float*)d_in[6];
  const float* wv_b  = (const float*)d_in[7];
  const float* scale = (const float*)d_in[8];
  (void)in_sizes; (void)n_in; (void)out_size; (void)ws_size;

  float* ws_knp = (float*)d_ws;            // 8*6*64*2   = 6144 floats
  float* ws_vp  = ws_knp + 6144;           // 8*32*192*2 = 98304 floats

  float* gout  = (float*)d_out;                       // [8,16384,192]
  float* gattn = gout + (size_t)8 * 16384 * 192;      // [8,16384,64]

  prek_kn<<<8, 64, 0, stream>>>(td, wk_w, wk_b, ws_knp);
  prek_v<<<96, 256, 0, stream>>>(td, wv_w, wv_b, ws_vp);
  attn_main<<<1024, 256, 103424, stream>>>(x, wq_w, wq_b, scale,
                                           ws_knp, ws_vp, gout, gattn);
}